// PatchReduction_overlap_72378788872306
// MI455X (gfx1250) — compile-verified
//
#include <hip/hip_runtime.h>
#include <hip/hip_bf16.h>

// PatchReduction_overlap: pure gather (0 FLOPs), bandwidth-bound.
// out[c,y,x] = in[(y/126)*9 + (x/126)][c][y%126][x%126]
// in : float32 [81][64][128][128]  (81 = 9x9 patches)
// out: float32 [64][1024][1024]
//
// Roofline: 256MB read + 256MB write => ~22us floor @ 23.3 TB/s.
// WMMA is inapplicable (no arithmetic); the win is coalesced b128 stores,
// coalesced b32 loads (contiguous within each 126-wide patch segment),
// and non-temporal cache hints since the 512MB stream exceeds the 192MB L2.

#define PR_STRIDE 126   // PATCH - OVERLAP
#define PR_NW     9
#define PR_PATCH  128
#define PR_C      64
#define PR_HW     1024

typedef float v4f __attribute__((ext_vector_type(4)));

__global__ __launch_bounds__(256) void
patch_gather_kernel(const float* __restrict__ in, float* __restrict__ out) {
    // blockIdx.x in [0, 64*1024): encodes (c, y)
    const unsigned y = blockIdx.x & (PR_HW - 1);
    const unsigned c = blockIdx.x >> 10;

    // Row-invariant terms (clamp in reference is a no-op for y < 1024).
    const unsigned iy   = y / PR_STRIDE;          // 0..8, v_mul_hi
    const unsigned offy = y - iy * PR_STRIDE;     // 0..125

    // Base offset into input for (patch row iy, channel c, patch-line offy).
    // patch stride = 64*128*128 = 1048576 elems; channel stride = 16384.
    const unsigned base = iy * (PR_NW * PR_C * PR_PATCH * PR_PATCH)
                        + c * (PR_PATCH * PR_PATCH)
                        + offy * PR_PATCH;

    const unsigned x0 = threadIdx.x * 4u;         // this thread's 4 x-values

    v4f v;
#pragma unroll
    for (int k = 0; k < 4; ++k) {
        const unsigned x    = x0 + (unsigned)k;
        const unsigned jx   = x / PR_STRIDE;      // 0..8
        const unsigned offx = x - jx * PR_STRIDE; // 0..125
        const unsigned idx  = base + jx * (PR_C * PR_PATCH * PR_PATCH) + offx;
        v[k] = __builtin_nontemporal_load(in + idx);
    }

    // out[c][y][x0..x0+3] — 16B-aligned contiguous store.
    v4f* dst = (v4f*)(out + ((size_t)c << 20) + ((size_t)y << 10) + x0);
    __builtin_nontemporal_store(v, dst);
}

extern "C" void kernel_launch(void* const* d_in, const int* in_sizes, int n_in,
                              void* d_out, int out_size, void* d_ws, size_t ws_size,
                              hipStream_t stream) {
    (void)in_sizes; (void)n_in; (void)d_ws; (void)ws_size; (void)out_size;
    const float* in = (const float*)d_in[0];
    float* out = (float*)d_out;

    // One block per (c, y) row: 64 * 1024 = 65536 blocks, 256 threads each
    // (8 wave32s); each thread writes one float4 of the output row.
    dim3 grid(PR_C * PR_HW);
    dim3 block(256);
    patch_gather_kernel<<<grid, block, 0, stream>>>(in, out);
}